// BridgeImportanceHGNN_23811298689762
// MI455X (gfx1250) — compile-verified
//
#include <hip/hip_runtime.h>
#include <math.h>

typedef __attribute__((ext_vector_type(2))) float v2f;
typedef __attribute__((ext_vector_type(8))) float v8f;

#define NODES 100000
#define EDGES 1250000
#define DIM   64
#define NTILES (NODES / 16)   // 6250

// ---------- float atomic max via int bit tricks (valid for mixed signs, init -inf)
__device__ __forceinline__ void atomicMaxF(float* addr, float val) {
    if (val >= 0.0f) {
        atomicMax((int*)addr, __float_as_int(val));
    } else {
        atomicMin((unsigned int*)addr, __float_as_uint(val));
    }
}

// ---------- K0: init agg(=d_out) to -inf, m to -inf, z to 0
__global__ void k0_init(float* agg_out, float* m, float* z) {
    int idx = blockIdx.x * blockDim.x + threadIdx.x;
    if (idx < NODES * DIM) agg_out[idx] = -INFINITY;
    if (idx < NODES) { m[idx] = -INFINITY; z[idx] = 0.0f; }
}

// ---------- K1: per-node attention projections a_i = x.w_src, a_j = x.w_dst
// one wave32 per node; lane loads 2 elements, cross-lane reduce via shfl_xor
__global__ void k1_attn(const float* __restrict__ x,
                        const float* __restrict__ w_src,
                        const float* __restrict__ w_dst,
                        float* __restrict__ a_i, float* __restrict__ a_j) {
    int node = blockIdx.x * 8 + (threadIdx.x >> 5);
    if (node >= NODES) return;
    int lane = threadIdx.x & 31;
    float x0 = x[node * DIM + lane];
    float x1 = x[node * DIM + lane + 32];
    float s = x0 * w_src[lane] + x1 * w_src[lane + 32];
    float t = x0 * w_dst[lane] + x1 * w_dst[lane + 32];
    #pragma unroll
    for (int msk = 16; msk >= 1; msk >>= 1) {
        s += __shfl_xor(s, msk, 32);
        t += __shfl_xor(t, msk, 32);
    }
    if (lane == 0) { a_i[node] = s; a_j[node] = t; }
}

__device__ __forceinline__ float lrelu(float v) {
    return v > 0.0f ? v : 0.2f * v;
}

// ---------- K2: segment max of leaky_relu(a_i[dst]+a_j[src]) over dst
__global__ void k2_segmax(const int* __restrict__ src, const int* __restrict__ dst,
                          const float* __restrict__ a_i, const float* __restrict__ a_j,
                          float* __restrict__ m) {
    int e = blockIdx.x * blockDim.x + threadIdx.x;
    if (e >= EDGES) return;
    int s = src[e], t = dst[e];
    float al = lrelu(a_i[t] + a_j[s]);
    atomicMaxF(&m[t], al);
}

// ---------- K3: z[dst] += exp(alpha - m[dst])
__global__ void k3_segsum(const int* __restrict__ src, const int* __restrict__ dst,
                          const float* __restrict__ a_i, const float* __restrict__ a_j,
                          const float* __restrict__ m, float* __restrict__ z) {
    int e = blockIdx.x * blockDim.x + threadIdx.x;
    if (e >= EDGES) return;
    int s = src[e], t = dst[e];
    float al = lrelu(a_i[t] + a_j[s]);
    atomicAdd(&z[t], __expf(al - m[t]));
}

// ---------- K4: agg[dst,d] = max over edges of (softmax weight * x[src,d])
// one thread per (edge, feature); broadcast scalars hit L2/WGP$
__global__ void k4_scatter(const int* __restrict__ src, const int* __restrict__ dst,
                           const float* __restrict__ a_i, const float* __restrict__ a_j,
                           const float* __restrict__ m, const float* __restrict__ z,
                           const float* __restrict__ x, float* __restrict__ agg_out) {
    int tid = blockIdx.x * blockDim.x + threadIdx.x;
    int e = tid >> 6;
    int d = tid & 63;
    if (e >= EDGES) return;
    int s = src[e], t = dst[e];
    float al = lrelu(a_i[t] + a_j[s]);
    float w = __expf(al - m[t]) / (z[t] + 1e-16f);
    float val = w * x[s * DIM + d];
    atomicMaxF(&agg_out[t * DIM + d], val);
}

// ---------- K5: out = agg @ W_l^T + b_l + x @ W_r^T   via V_WMMA_F32_16X16X4_F32
// agg is aliased in `out`: each wave reads all A-tiles of its 16 rows (for all
// 4 column tiles) BEFORE any store to those rows -> no aliasing hazard.
// A layout (16x4 f32): lanes 0-15 hold M=lane K={k0,k0+1}; lanes 16-31 M=lane-16 K={k0+2,k0+3}
// B layout (4x16 f32): lanes 0-15 hold N=lane K={0,1}; lanes 16-31 N=lane-16 K={2,3}
// C/D layout: VGPR j -> row j (lanes 0-15, col=lane) / row j+8 (lanes 16-31, col=lane-16)
__global__ void k5_gemm(const float* __restrict__ x,
                        const float* __restrict__ W_l,
                        const float* __restrict__ b_l,
                        const float* __restrict__ W_r,
                        float* out /* holds agg on entry */) {
    int tile = blockIdx.x * 8 + (threadIdx.x >> 5);
    if (tile >= NTILES) return;          // wave-uniform: EXEC stays all-ones for WMMA
    int lane  = threadIdx.x & 31;
    int row0  = tile * 16;
    int mloc  = lane & 15;
    int khalf = (lane >> 4) * 2;         // 0 for lanes 0-15, 2 for lanes 16-31

    v8f acc[4];
    #pragma unroll
    for (int ct = 0; ct < 4; ++ct) acc[ct] = (v8f){0,0,0,0,0,0,0,0};

    #pragma unroll
    for (int ct = 0; ct < 4; ++ct) {
        int ocol = ct * 16 + mloc;       // output column this lane supplies for B
        #pragma unroll
        for (int k0 = 0; k0 < DIM; k0 += 4) {
            int kA = k0 + khalf;
            const int arow = (row0 + mloc) * DIM + kA;
            float g0 = out[arow];
            float g1 = out[arow + 1];
            v2f a_agg, a_x, b_wl, b_wr;
            a_agg.x = (g0 == -INFINITY) ? 0.0f : g0;   // PyG: empty-segment max -> 0
            a_agg.y = (g1 == -INFINITY) ? 0.0f : g1;
            a_x.x = x[arow];
            a_x.y = x[arow + 1];
            // B[k][o] = W[o][k]  (W stored [out,in] row-major)
            b_wl.x = W_l[ocol * DIM + kA];
            b_wl.y = W_l[ocol * DIM + kA + 1];
            b_wr.x = W_r[ocol * DIM + kA];
            b_wr.y = W_r[ocol * DIM + kA + 1];
            acc[ct] = __builtin_amdgcn_wmma_f32_16x16x4_f32(
                false, a_agg, false, b_wl, (short)0, acc[ct], false, false);
            acc[ct] = __builtin_amdgcn_wmma_f32_16x16x4_f32(
                false, a_x,   false, b_wr, (short)0, acc[ct], false, false);
        }
    }

    // all reads of agg rows complete -> safe to overwrite with final output
    int rbase = row0 + ((lane >> 4) ? 8 : 0);
    int cloc  = lane & 15;
    #pragma unroll
    for (int ct = 0; ct < 4; ++ct) {
        int col = ct * 16 + cloc;
        float bias = b_l[col];
        #pragma unroll
        for (int j = 0; j < 8; ++j) {
            out[(rbase + j) * DIM + col] = acc[ct][j] + bias;
        }
    }
}

extern "C" void kernel_launch(void* const* d_in, const int* in_sizes, int n_in,
                              void* d_out, int out_size, void* d_ws, size_t ws_size,
                              hipStream_t stream) {
    const float* x      = (const float*)d_in[0];
    const int*   ei     = (const int*)d_in[1];      // [2, E] flat
    const float* W_l    = (const float*)d_in[2];
    const float* b_l    = (const float*)d_in[3];
    const float* W_r    = (const float*)d_in[4];
    const float* w_src  = (const float*)d_in[5];    // att_src
    const float* w_dst  = (const float*)d_in[6];    // att_dst
    float* out = (float*)d_out;

    const int* src = ei;            // edge_index[0]
    const int* dst = ei + EDGES;    // edge_index[1]

    // workspace: m | z | a_i | a_j   (4 * N floats = 1.6 MB)
    float* m   = (float*)d_ws;
    float* z   = m + NODES;
    float* a_i = z + NODES;
    float* a_j = a_i + NODES;

    const int T = 256;
    // K0: init agg(=out) / m / z
    k0_init<<<(NODES * DIM + T - 1) / T, T, 0, stream>>>(out, m, z);
    // K1: per-node attention dot products (one wave per node)
    k1_attn<<<(NODES + 7) / 8, T, 0, stream>>>(x, w_src, w_dst, a_i, a_j);
    // K2: segment max of attention logits
    k2_segmax<<<(EDGES + T - 1) / T, T, 0, stream>>>(src, dst, a_i, a_j, m);
    // K3: segment sum of exp
    k3_segsum<<<(EDGES + T - 1) / T, T, 0, stream>>>(src, dst, a_i, a_j, m, z);
    // K4: attention-weighted scatter-max of features
    k4_scatter<<<(EDGES * 64 + T - 1) / T, T, 0, stream>>>(src, dst, a_i, a_j, m, z, x, out);
    // K5: fused dual GEMM + bias via fp32 WMMA (one wave per 16-row tile)
    k5_gemm<<<(NTILES + 7) / 8, T, 0, stream>>>(x, W_l, b_l, W_r, out);
}